// N3Block_29841432773337
// MI455X (gfx1250) — compile-verified
//
#include <hip/hip_runtime.h>
#include <math.h>

typedef __attribute__((ext_vector_type(2))) float v2f;
typedef __attribute__((ext_vector_type(8))) float v8f;

#define B_ 2
#define N_ 4096
#define M_ 4096
#define O_ 48
#define E_ 64
#define F_ 256
#define K_ 7
#define XNB_STRIDE 272           // F_ + 16: lane-halves (delta o = 2) land on disjoint LDS banks
#define W_STRIDE 16              // k-stride of s_W; cols 7..15 are zero padding (no guarded loads)
#define Z_STRIDE 16              // k-stride of s_z; cols 7..15 scratch (no guarded stores)
#define LOG_HALF_F (-0.6931471805599453f)

__device__ __forceinline__ float log1mexp_f(float x) {
    // matches reference: x < log(0.5) -> log1p(-exp(x)), else log(-expm1(x))
    return (x < LOG_HALF_F) ? log1pf(-expf(x)) : logf(-expm1f(x));
}

__global__ __launch_bounds__(256) void n3block_kernel(
    const float* __restrict__ x,
    const float* __restrict__ xe,
    const float* __restrict__ ye,
    const int*   __restrict__ I,
    const float* __restrict__ log_temp,
    float* __restrict__ out)
{
    __shared__ float s_xnb[O_ * XNB_STRIDE];   // gathered payload rows (52224 B)
    __shared__ float s_ye[E_];
    __shared__ int   s_idx[O_];
    __shared__ float s_D[O_];                  // 2*dot - |xe_o|^2
    __shared__ float s_W[O_ * W_STRIDE];       // [o][k], zero-padded cols 7..15
    __shared__ float s_z[F_ * Z_STRIDE];       // [f][k], cols 7..15 scratch

    const int tid  = threadIdx.x;
    const int q    = blockIdx.x;               // flat (b, m)
    const int b    = q / M_;
    const int lane = tid & 31;
    const int wv   = tid >> 5;

    const float* xe_b = xe + (size_t)b * N_ * E_;
    const float* x_b  = x  + (size_t)b * N_ * F_;

    // ---- stage 0: indices + query embedding into LDS; zero-fill W padding ----
    if (tid < O_)                    s_idx[tid]     = I[(size_t)q * O_ + tid];
    if (tid >= 64 && tid < 64 + E_)  s_ye[tid - 64] = ye[(size_t)q * E_ + (tid - 64)];
    #pragma unroll
    for (int i = tid; i < O_ * W_STRIDE; i += 256) s_W[i] = 0.f;
    __syncthreads();

    // ---- stage 1b (issued first): async DMA gather of 48 payload rows into LDS ----
    // GLOBAL_LOAD_ASYNC_TO_LDS_B128, GVS mode: mem = SGPR64(x_b) + VGPR32(byte off).
    // Tracked by ASYNCcnt, so it overlaps the distance computation below.
    {
        const int orow = tid >> 6;             // 0..3
        const int f4b  = (tid & 63) * 16;      // byte offset of this lane's 16B chunk
        #pragma unroll
        for (int g = 0; g < O_ / 4; ++g) {
            const int o = g * 4 + orow;
            unsigned vaddr  = (unsigned)s_idx[o] * (unsigned)(F_ * 4) + (unsigned)f4b;
            unsigned ldsoff = (unsigned)(size_t)&s_xnb[o * XNB_STRIDE] + (unsigned)f4b;
            asm volatile("global_load_async_to_lds_b128 %0, %1, %2"
                         :: "v"(ldsoff), "v"(vaddr), "s"(x_b)
                         : "memory");
        }
    }

    // ---- stage 1a: distances, 4 lanes per neighbor (overlaps async DMA) ----
    if (tid < O_ * 4) {
        const int o  = tid >> 2;
        const int sq = tid & 3;
        const float4* row = (const float4*)(xe_b + (size_t)s_idx[o] * E_);
        float dot = 0.f, nrm = 0.f;
        #pragma unroll
        for (int j = 0; j < 4; ++j) {
            float4 v = row[sq * 4 + j];
            const int e = sq * 16 + j * 4;
            dot += v.x * s_ye[e] + v.y * s_ye[e + 1] + v.z * s_ye[e + 2] + v.w * s_ye[e + 3];
            nrm += v.x * v.x + v.y * v.y + v.z * v.z + v.w * v.w;
        }
        dot += __shfl_xor(dot, 1, 32); dot += __shfl_xor(dot, 2, 32);
        nrm += __shfl_xor(nrm, 1, 32); nrm += __shfl_xor(nrm, 2, 32);
        if (sq == 0) s_D[o] = 2.f * dot - nrm;
    }

    asm volatile("s_wait_asynccnt 0" ::: "memory");   // this wave's DMA done
    __syncthreads();                                   // all waves' DMA + D[o] visible

    // ---- stage 2: continuous top-K weights (wave 0; lanes cover o and o+32) ----
    if (wv == 0) {
        float y0 = s_ye[lane], y1 = s_ye[lane + 32];
        float ysq = y0 * y0 + y1 * y1;
        #pragma unroll
        for (int off = 16; off >= 1; off >>= 1) ysq += __shfl_xor(ysq, off, 32);
        const float inv_t = expf(-log_temp[0]);            // 1/temperature, LOG_TEMP_BIAS = 0
        float lg0 = (s_D[lane] - ysq) * inv_t;
        float lg1 = (lane < 16) ? (s_D[lane + 32] - ysq) * inv_t : -1e30f;
        #pragma unroll
        for (int k = 0; k < K_; ++k) {
            float mx = fmaxf(lg0, lg1);
            #pragma unroll
            for (int off = 16; off >= 1; off >>= 1) mx = fmaxf(mx, __shfl_xor(mx, off, 32));
            float e0 = expf(lg0 - mx);
            float e1 = (lane < 16) ? expf(lg1 - mx) : 0.f;
            float sm = e0 + e1;
            #pragma unroll
            for (int off = 16; off >= 1; off >>= 1) sm += __shfl_xor(sm, off, 32);
            const float ls = mx + logf(sm);
            const float w0 = lg0 - ls;                     // log-softmax value
            s_W[lane * W_STRIDE + k] = expf(w0);
            lg0 += log1mexp_f(w0);
            if (lane < 16) {
                const float w1 = lg1 - ls;
                s_W[(lane + 32) * W_STRIDE + k] = expf(w1);
                lg1 += log1mexp_f(w1);
            }
        }
    }
    __syncthreads();

    // ---- stage 3: z[f,k] = sum_o x_nb[o,f] * W[o,k] via V_WMMA_F32_16X16X4_F32 ----
    {
        const int mrow = lane & 15;            // A row / B & D column
        const int half = lane >> 4;            // K-pair select: {0,1} vs {2,3}

        // B fragments are f-tile independent: load once, reuse across both tiles.
        v2f bmat[O_ / 4];
        const float* bp = &s_W[(half * 2) * W_STRIDE + mrow];
        #pragma unroll
        for (int g = 0; g < O_ / 4; ++g) {
            bmat[g][0] = bp[g * 4 * W_STRIDE];                 // W[4g + 2*half    ][mrow]
            bmat[g][1] = bp[g * 4 * W_STRIDE + W_STRIDE];      // W[4g + 2*half + 1][mrow]
        }

        #pragma unroll
        for (int ft = 0; ft < 2; ++ft) {
            const int fbase = (wv + ft * 8) * 16;
            const float* ap = &s_xnb[(half * 2) * XNB_STRIDE + fbase + mrow];
            v8f acc = {0.f, 0.f, 0.f, 0.f, 0.f, 0.f, 0.f, 0.f};
            #pragma unroll
            for (int g = 0; g < O_ / 4; ++g) {
                v2f a;
                a[0] = ap[g * 4 * XNB_STRIDE];                 // x_nb[4g + 2*half    ][f]
                a[1] = ap[g * 4 * XNB_STRIDE + XNB_STRIDE];    // x_nb[4g + 2*half + 1][f]
                acc = __builtin_amdgcn_wmma_f32_16x16x4_f32(
                        false, a, false, bmat[g], (short)0, acc, false, false);
            }
            // D: lane l, vgpr v -> (f = fbase + v + 8*half, k = l&15); cols >=7 land in padding
            float* zp = &s_z[(fbase + half * 8) * Z_STRIDE + mrow];
            #pragma unroll
            for (int v = 0; v < 8; ++v) zp[v * Z_STRIDE] = acc[v];
        }
    }
    __syncthreads();

    // ---- stage 4: coalesced store of (F,K) slab; F_*K_ = 1792 = 7*256 ----
    {
        float* dst = out + (size_t)q * (F_ * K_);
        #pragma unroll
        for (int it = 0; it < 7; ++it) {
            const int i = it * 256 + tid;
            const int f = i / K_;
            const int k = i - f * K_;
            dst[i] = s_z[f * Z_STRIDE + k];
        }
    }
}

extern "C" void kernel_launch(void* const* d_in, const int* in_sizes, int n_in,
                              void* d_out, int out_size, void* d_ws, size_t ws_size,
                              hipStream_t stream) {
    const float* x        = (const float*)d_in[0];
    const float* xe       = (const float*)d_in[1];
    const float* ye       = (const float*)d_in[2];
    const int*   I        = (const int*)d_in[3];
    const float* log_temp = (const float*)d_in[4];
    float* out            = (float*)d_out;

    dim3 grid(B_ * M_);   // one workgroup per query (b,m)
    dim3 block(256);      // 8 wave32
    hipLaunchKernelGGL(n3block_kernel, grid, block, 0, stream,
                       x, xe, ye, I, log_temp, out);
}